// GraphConv_15401752724058
// MI455X (gfx1250) — compile-verified
//
#include <hip/hip_runtime.h>

typedef __attribute__((ext_vector_type(2))) float v2f;
typedef __attribute__((ext_vector_type(8))) float v8f;

#define NNODES 50000
#define DDIM   128

// ---------------------------------------------------------------------------
// Kernel 1: h = seq @ W.T  via V_WMMA_F32_16X16X4_F32 (true fp32 matrix pipe)
// Block = 256 threads = 8 waves. Each wave owns TWO 16x16 output tiles that
// share the same 16 output columns (j0 = 16*waveId) but cover two row tiles
// (n0, n0+16): two independent accumulator chains hide WMMA pipeline latency,
// and the B operand (W tile) is loaded once per k-step and fed to both WMMAs.
//
// Operand layouts (ISA 7.12.2, 32-bit):
//   A 16x4 : lane L holds A[L&15][2*(L>>4) + v], v=0,1  -> contiguous float2
//   B 4x16 : lane L holds B[v + 2*(L>>4)][L&15]         -> B[K][N] = W[j0+N][k],
//            so lane L reads W row (j0 + (L&15)) at k + 2*(L>>4): contiguous float2
//   C/D    : lane L, vgpr v holds D[v + 8*(L>>4)][L&15]
// ---------------------------------------------------------------------------
__global__ __launch_bounds__(256) void gemm_wmma_f32(
    const float* __restrict__ seq, const float* __restrict__ Wm,
    float* __restrict__ h) {
  const int wave = threadIdx.x >> 5;
  const int lane = threadIdx.x & 31;
  const int half = lane >> 4;   // 0 or 1
  const int idx  = lane & 15;
  const int n0 = blockIdx.x * 32;   // two 16-row tiles per block
  const int j0 = wave * 16;

  const int r0 = n0 + idx;                       // tile-0 row: always < NNODES
  int r1 = n0 + 16 + idx;                        // tile-1 row: may overrun
  if (r1 >= NNODES) r1 = NNODES - 1;             // clamp (dup read, stores guarded)

  const float* aRow0 = seq + (size_t)r0 * DDIM + 2 * half;
  const float* aRow1 = seq + (size_t)r1 * DDIM + 2 * half;
  const float* bRow  = Wm  + (size_t)(j0 + idx) * DDIM + 2 * half;

  v8f c0 = {};
  v8f c1 = {};
#pragma unroll
  for (int k = 0; k < DDIM; k += 4) {
    v2f b  = *(const v2f*)(bRow + k);     // shared by both row tiles
    v2f a0 = *(const v2f*)(aRow0 + k);
    v2f a1 = *(const v2f*)(aRow1 + k);
    c0 = __builtin_amdgcn_wmma_f32_16x16x4_f32(
        false, a0, false, b, (short)0, c0, false, false);
    c1 = __builtin_amdgcn_wmma_f32_16x16x4_f32(
        false, a1, false, b, (short)0, c1, false, false);
  }

#pragma unroll
  for (int v = 0; v < 8; ++v) {
    const int m0 = n0 + v + 8 * half;            // always < NNODES
    h[(size_t)m0 * DDIM + j0 + idx] = c0[v];
    const int m1 = n0 + 16 + v + 8 * half;
    if (m1 < NNODES) h[(size_t)m1 * DDIM + j0 + idx] = c1[v];
  }
}

// ---------------------------------------------------------------------------
// Kernel 2: COO scatter: out[row] += val * h[col].  One wave32 per edge,
// each lane moves a float4 chunk (32 lanes x 16B = 512B = one row of h).
// h (25.6 MB) is L2-resident on MI455X (192 MB L2); fp32 hardware atomics
// (global_atomic_add_f32) resolve in L2.
// ---------------------------------------------------------------------------
__global__ __launch_bounds__(256) void spmm_scatter(
    const float* __restrict__ h, const int* __restrict__ erow,
    const int* __restrict__ ecol, const float* __restrict__ eval,
    float* __restrict__ out, int n_edges) {
  const long gid = (long)blockIdx.x * blockDim.x + threadIdx.x;
  const int e = (int)(gid >> 5);
  const int lane = (int)(gid & 31);
  if (e >= n_edges) return;

  const int row = erow[e];
  const int col = ecol[e];
  const float val = eval[e];

  const float4 m = ((const float4*)(h + (size_t)col * DDIM))[lane];
  float* dst = out + (size_t)row * DDIM + lane * 4;
  unsafeAtomicAdd(dst + 0, m.x * val);
  unsafeAtomicAdd(dst + 1, m.y * val);
  unsafeAtomicAdd(dst + 2, m.z * val);
  unsafeAtomicAdd(dst + 3, m.w * val);
}

// ---------------------------------------------------------------------------
// Kernel 0: zero-init the accumulation buffer (harness poisons d_out).
// ---------------------------------------------------------------------------
__global__ __launch_bounds__(256) void zero_f32(float* __restrict__ p, int n) {
  int i = blockIdx.x * blockDim.x + threadIdx.x;
  if (i < n) p[i] = 0.0f;
}

// ---------------------------------------------------------------------------
// Kernel 3: fused epilogue out = relu(out + b)
// ---------------------------------------------------------------------------
__global__ __launch_bounds__(256) void bias_relu(
    float* __restrict__ out, const float* __restrict__ b, int n) {
  int i = blockIdx.x * blockDim.x + threadIdx.x;
  if (i < n) {
    float v = out[i] + b[i & (DDIM - 1)];
    out[i] = v > 0.0f ? v : 0.0f;
  }
}

extern "C" void kernel_launch(void* const* d_in, const int* in_sizes, int n_in,
                              void* d_out, int out_size, void* d_ws, size_t ws_size,
                              hipStream_t stream) {
  const float* seq  = (const float*)d_in[0];
  const float* Wm   = (const float*)d_in[1];
  const float* b    = (const float*)d_in[2];
  const int*   erow = (const int*)d_in[3];
  const int*   ecol = (const int*)d_in[4];
  const float* eval = (const float*)d_in[5];
  float* out = (float*)d_out;
  float* h   = (float*)d_ws;  // 50000*128*4 = 25.6 MB scratch

  const int n_edges = in_sizes[3];
  const int total = NNODES * DDIM;

  zero_f32<<<(total + 255) / 256, 256, 0, stream>>>(out, total);
  gemm_wmma_f32<<<(NNODES + 31) / 32, 256, 0, stream>>>(seq, Wm, h);
  const long sthreads = (long)n_edges * 32;
  spmm_scatter<<<(int)((sthreads + 255) / 256), 256, 0, stream>>>(
      h, erow, ecol, eval, out, n_edges);
  bias_relu<<<(total + 255) / 256, 256, 0, stream>>>(out, b, total);
}